// Model_5454608466608
// MI455X (gfx1250) — compile-verified
//
#include <hip/hip_runtime.h>
#include <hip/hip_bf16.h>

// ---------------------------------------------------------------------------
// CDNA5 (gfx1250) mesh-ops kernel: verts+center, 4x replication, two SpMV
// losses. Memory-bound (~200MB @ 23.3 TB/s ~= 9us floor).
//  - WMMA f32 16x16x4 as the wave32 reduction engine for scalar losses.
//  - Tensor Data Mover (tensor_load_to_lds / tensor_store_from_lds) for the
//    faces replication: DMA-staged through LDS, no VGPR/VALU traffic.
// ---------------------------------------------------------------------------

typedef __attribute__((ext_vector_type(2))) float        v2f;
typedef __attribute__((ext_vector_type(8))) float        v8f;
typedef __attribute__((ext_vector_type(4))) int          v4i;
typedef __attribute__((ext_vector_type(4))) unsigned int v4u;
typedef __attribute__((ext_vector_type(8))) unsigned int v8u;

#if __has_builtin(__builtin_amdgcn_wmma_f32_16x16x4_f32)
#define HAVE_WMMA_X4 1
#else
#define HAVE_WMMA_X4 0
#pragma message("PROBE: __builtin_amdgcn_wmma_f32_16x16x4_f32 MISSING")
#endif

// ---------------------------------------------------------------------------
// Wave32 sum using V_WMMA_F32_16X16X4_F32 with B = ones.
// A layout (16x4 f32): lanes 0-15 supply (K0,K1), lanes 16-31 supply (K2,K3),
// M = lane%16.  With a = {s, 0}:  D[m][n] = s_m + s_{m+16} for every n.
// Summing the 8 D VGPRs gives half-wave sums; shfl_xor(16) combines halves.
// EXEC must be all ones at the call site.
// ---------------------------------------------------------------------------
__device__ __forceinline__ float wave_sum32(float s) {
#if HAVE_WMMA_X4
  v2f a; a[0] = s;   a[1] = 0.0f;
  v2f b; b[0] = 1.0f; b[1] = 1.0f;
  v8f c = {};
  c = __builtin_amdgcn_wmma_f32_16x16x4_f32(
      /*neg_a=*/false, a, /*neg_b=*/false, b,
      /*c_mod=*/(short)0, c, /*reuse_a=*/false, /*reuse_b=*/false);
  float t = ((c[0] + c[1]) + (c[2] + c[3])) + ((c[4] + c[5]) + (c[6] + c[7]));
  t += __shfl_xor(t, 16, 32);
  return t;
#else
  for (int off = 16; off > 0; off >>= 1) s += __shfl_xor(s, off, 32);
  return s;
#endif
}

// ---------------------------------------------------------------------------
// K1: zero the Laplacian accumulator (V*3 floats in d_ws) and the 4 scalar
// output slots.  Must run every launch (graph replays don't re-poison).
// ---------------------------------------------------------------------------
__global__ void k_init(float* __restrict__ ws, int v3, float* __restrict__ sc) {
  int i = blockIdx.x * blockDim.x + threadIdx.x;
  if (i < v3) ws[i] = 0.0f;
  if (i < 4) sc[i] = 0.0f;
}

// ---------------------------------------------------------------------------
// K2: verts = vertices + center; copy 0 with regular stores (re-read by the
// SpMV kernels -> keep in L2), copies 1..R-1 non-temporal streaming.
// ---------------------------------------------------------------------------
__global__ void k_verts_rep(const float* __restrict__ vtx,
                            const float* __restrict__ ctr,
                            float* __restrict__ out, int v3, int reps) {
  int i = blockIdx.x * blockDim.x + threadIdx.x;
  if (i >= v3) return;
  float v = vtx[i] + ctr[(unsigned)i % 3u];
  out[i] = v;
  for (int c = 1; c < reps; ++c)
    __builtin_nontemporal_store(v, out + (size_t)c * v3 + i);
}

// ---------------------------------------------------------------------------
// K3: faces_rep via the Tensor Data Mover.  One wave per 32KB tile:
//   tensor_load_to_lds (tile -> LDS), s_wait_tensorcnt 0,
//   R x tensor_store_from_lds (LDS -> each replica).
// Tail handled by hardware range-check: tensor_dim0 = remaining elements, so
// OOB loads return 0 and OOB stores are dropped.
// ---------------------------------------------------------------------------
#define TDM_TILE 8192   // int32 elements per tile = 32 KB LDS

// Build D# group0 (4 dwords) + group1 (8 dwords) for a 1-D 4-byte-element tile.
__device__ __forceinline__ void tdm_desc(unsigned long long gaddr,
                                         unsigned int lds_byte,
                                         unsigned int valid_elems,  // tensor_dim0
                                         unsigned int tile_elems,   // tile_dim0
                                         v4u& g0, v8u& g1) {
  g0[0] = 1u;                                                // count=1 (valid), user desc
  g0[1] = lds_byte;                                          // lds_addr
  g0[2] = (unsigned int)gaddr;                               // global_addr[31:0]
  g0[3] = ((unsigned int)(gaddr >> 32) & 0x01FFFFFFu)        // global_addr[56:32]
          | (2u << 30);                                      // type = 2 ("image")
  g1[0] = (2u << 16);                                        // data_size=2 (4B); mask/flags=0
  g1[1] = (valid_elems & 0xFFFFu) << 16;                     // tensor_dim0[15:0]
  g1[2] = (valid_elems >> 16) | (1u << 16);                  // tensor_dim0[31:16] | tensor_dim1=1
  g1[3] = (tile_elems & 0xFFFFu) << 16;                      // tensor_dim1 hi=0 | tile_dim0
  g1[4] = 0u;                                                // tile_dim1=0 (1-D), tile_dim2=0
  g1[5] = valid_elems;                                       // tensor_dim0_stride[31:0]
  g1[6] = (valid_elems >> 16) & 0xFFFFu;                     // stride0[47:32] | stride1 lo=0
  g1[7] = 0u;                                                // stride1 hi
}

__global__ void __launch_bounds__(32)
k_faces_rep_tdm(const int* __restrict__ faces, int* __restrict__ out,
                int n, int reps) {
  __shared__ __attribute__((aligned(128))) int buf[TDM_TILE];
  long start = (long)blockIdx.x * TDM_TILE;
  if (start >= n) return;
  unsigned int remaining = (unsigned int)(n - start);

  unsigned int lds_byte = (unsigned int)(unsigned long long)
      (__attribute__((address_space(3))) int*)buf;

  v4u gz; gz[0] = 0u; gz[1] = 0u; gz[2] = 0u; gz[3] = 0u;

  // Load tile -> LDS via TDM.
  {
    v4u g0; v8u g1;
    tdm_desc((unsigned long long)(uintptr_t)(faces + start),
             lds_byte, remaining, TDM_TILE, g0, g1);
    asm volatile("tensor_load_to_lds %0, %1, %2, %3"
                 :: "s"(g0), "s"(g1), "s"(gz), "s"(gz) : "memory");
  }
  asm volatile("s_wait_tensorcnt 0x0" ::: "memory");

  // Store tile from LDS to each replica via TDM.
  for (int c = 0; c < reps; ++c) {
    v4u g0; v8u g1;
    tdm_desc((unsigned long long)(uintptr_t)(out + (size_t)c * n + start),
             lds_byte, remaining, TDM_TILE, g0, g1);
    asm volatile("tensor_store_from_lds %0, %1, %2, %3"
                 :: "s"(g0), "s"(g1), "s"(gz), "s"(gz) : "memory");
  }
  asm volatile("s_wait_tensorcnt 0x0" ::: "memory");
}

// ---------------------------------------------------------------------------
// K4: Laplacian scatter.  Rows are sorted (np.unique of row*V+col) except for
// the appended diagonal block, so consecutive-equal-row merging inside each
// 8-nnz chunk cuts the f32 atomic count ~8x.  Atomics land in L2 (the 6 MB
// accumulator is L2-resident).
// ---------------------------------------------------------------------------
__global__ void k_lap_scatter(const int* __restrict__ rows,
                              const int* __restrict__ cols,
                              const float* __restrict__ vals,
                              const float* __restrict__ verts,
                              float* __restrict__ acc, int nnz) {
  const int CHUNK = 8;
  long base = ((long)blockIdx.x * blockDim.x + threadIdx.x) * CHUNK;
  if (base >= nnz) return;
  long end = base + CHUNK;
  if (end > nnz) end = nnz;
  int cur = rows[base];
  float ax = 0.0f, ay = 0.0f, az = 0.0f;
  for (long i = base; i < end; ++i) {
    int r = rows[i];
    if (r != cur) {
      atomicAdd(acc + 3ll * cur + 0, ax);
      atomicAdd(acc + 3ll * cur + 1, ay);
      atomicAdd(acc + 3ll * cur + 2, az);
      cur = r; ax = ay = az = 0.0f;
    }
    int c = cols[i];
    float v = vals[i];
    const float* p = verts + 3ll * c;
    ax += v * p[0]; ay += v * p[1]; az += v * p[2];
  }
  atomicAdd(acc + 3ll * cur + 0, ax);
  atomicAdd(acc + 3ll * cur + 1, ay);
  atomicAdd(acc + 3ll * cur + 2, az);
}

// ---------------------------------------------------------------------------
// K5: hexagon loss, fully fused: k_rows = repeat(arange(V), KD) means vertex i
// owns nnz [i*KD, (i+1)*KD).  Gather, dot, square, WMMA wave reduction.
// No early return before wave_sum32 (EXEC must be all ones for WMMA).
// ---------------------------------------------------------------------------
__global__ void k_hexagon(const int* __restrict__ kc, const float* __restrict__ kv,
                          const float* __restrict__ verts, int V, int KD,
                          float invV, float* __restrict__ out_acc) {
  int i = blockIdx.x * blockDim.x + threadIdx.x;
  float d = 0.0f;
  if (i < V) {
    float ax = 0.0f, ay = 0.0f, az = 0.0f;
    long base = (long)i * KD;
    for (int j = 0; j < KD; ++j) {
      int c = kc[base + j];
      float v = kv[base + j];
      const float* p = verts + 3ll * c;
      ax += v * p[0]; ay += v * p[1]; az += v * p[2];
    }
    d = ax * ax + ay * ay + az * az;
  }
  float w = wave_sum32(d * invV);
  if ((threadIdx.x & 31) == 0) atomicAdd(out_acc, w);
}

// ---------------------------------------------------------------------------
// K6: Laplacian loss: per-vertex L2 norm of the accumulator, WMMA-reduced.
// ---------------------------------------------------------------------------
__global__ void k_lap_norm(const float* __restrict__ acc, int V, float invV,
                           float* __restrict__ out_acc) {
  int i = blockIdx.x * blockDim.x + threadIdx.x;
  float nrm = 0.0f;
  if (i < V) {
    float x = acc[3ll * i + 0];
    float y = acc[3ll * i + 1];
    float z = acc[3ll * i + 2];
    nrm = sqrtf(x * x + y * y + z * z);
  }
  float w = wave_sum32(nrm * invV);
  if ((threadIdx.x & 31) == 0) atomicAdd(out_acc, w);
}

// ---------------------------------------------------------------------------
// Host launcher.  Input order (setup_inputs): vertices, center, lap_rows,
// lap_cols, lap_vals, k_rows, k_cols, k_vals, faces, total_num.
// Output layout (flat): vertices_rep [R*V*3] | faces_rep [R*NF*3] |
// lap_loss | hex_loss | 0 | 0.
// ---------------------------------------------------------------------------
extern "C" void kernel_launch(void* const* d_in, const int* in_sizes, int n_in,
                              void* d_out, int out_size, void* d_ws, size_t ws_size,
                              hipStream_t stream) {
  const float* vertices = (const float*)d_in[0];
  const float* center   = (const float*)d_in[1];
  const int*   lap_rows = (const int*)d_in[2];
  const int*   lap_cols = (const int*)d_in[3];
  const float* lap_vals = (const float*)d_in[4];
  const int*   k_cols   = (const int*)d_in[6];
  const float* k_vals   = (const float*)d_in[7];
  const int*   faces    = (const int*)d_in[8];

  const int V3  = in_sizes[0];          // V * 3
  const int V   = V3 / 3;
  const int nnz = in_sizes[2];
  const int KD  = in_sizes[5] / V;      // 7
  const int NF3 = in_sizes[8];          // NF * 3
  const int R   = (out_size - 4) / (V3 + NF3);   // total_num (== 4)

  float* out        = (float*)d_out;
  float* out_faces  = out + (size_t)R * V3;      // int bits stored via int*
  float* out_scalar = out + (size_t)R * (V3 + NF3);
  float* lap_acc    = (float*)d_ws;              // V*3 floats (6 MB)

  const float invV = 1.0f / (float)V;
  const dim3 B(256);

  // K1: zero accumulator + the 4 scalar slots.
  {
    int g = (V3 + 255) / 256;
    k_init<<<g, B, 0, stream>>>(lap_acc, V3, out_scalar);
  }
  // K2: verts + center, replicated R times (copy 0 feeds the SpMVs).
  {
    int g = (V3 + 255) / 256;
    k_verts_rep<<<g, B, 0, stream>>>(vertices, center, out, V3, R);
  }
  // K3: faces replicated R times through the Tensor Data Mover.
  {
    int g = (NF3 + TDM_TILE - 1) / TDM_TILE;
    if (g < 1) g = 1;
    k_faces_rep_tdm<<<g, dim3(32), 0, stream>>>(faces, (int*)out_faces, NF3, R);
  }
  // K4: Laplacian scatter-add into d_ws.
  {
    long threads = ((long)nnz + 7) / 8;
    int g = (int)((threads + 255) / 256);
    k_lap_scatter<<<g, B, 0, stream>>>(lap_rows, lap_cols, lap_vals,
                                       out /*verts copy 0*/, lap_acc, nnz);
  }
  // K5: hexagon loss (fused SpMV + squared-norm + WMMA reduction).
  {
    int g = (V + 255) / 256;
    k_hexagon<<<g, B, 0, stream>>>(k_cols, k_vals, out /*verts*/, V, KD,
                                   invV, out_scalar + 1);
  }
  // K6: Laplacian loss (norms of accumulator + WMMA reduction).
  {
    int g = (V + 255) / 256;
    k_lap_norm<<<g, B, 0, stream>>>(lap_acc, V, invV, out_scalar + 0);
  }
}